// CBOWModel_54211077210250
// MI455X (gfx1250) — compile-verified
//
#include <hip/hip_runtime.h>
#include <hip/hip_bf16.h>
#include <cstddef>

typedef float v2f __attribute__((ext_vector_type(2)));
typedef float v8f __attribute__((ext_vector_type(8)));

static constexpr int BATCH = 2048;
static constexpr int CTX   = 10;
static constexpr int EMBED = 128;
static constexpr int VOCAB = 100000;
static constexpr int NTILE = VOCAB / 16;   // 6250 column tiles of 16
static constexpr int SEG   = 8;            // column segments per row group
static constexpr int WPB   = 8;            // waves per block (256 threads, wave32)
static constexpr int TSTRIDE = SEG * WPB;  // tile stride across a row group
static constexpr float NEG_HUGE = -3.402823466e38f;

// ---------------------------------------------------------------------------
// Kernel 1: hidden[b][d] = mean_c W1[inputs[b][c]][d]
// ---------------------------------------------------------------------------
__global__ void cbow_hidden_kernel(const int* __restrict__ inputs,
                                   const float* __restrict__ W1,
                                   float* __restrict__ hidden) {
    int idx = blockIdx.x * blockDim.x + threadIdx.x;   // 2048*128 threads
    int b = idx >> 7;
    int d = idx & 127;
    const int* ip = inputs + b * CTX;
    float acc = 0.0f;
#pragma unroll
    for (int c = 0; c < CTX; ++c) {
        acc += W1[(size_t)ip[c] * EMBED + d];
    }
    hidden[idx] = acc * (1.0f / (float)CTX);
}

// ---------------------------------------------------------------------------
// Kernel 2: logits tile GEMM via V_WMMA_F32_16X16X4_F32 + online softmax stats
// grid = (BATCH/16) * SEG blocks, 256 threads (8 waves)
// B preloaded in two 16-step chunks (caps live registers, avoids spills),
// next tile prefetched into cache one iteration ahead.
// ---------------------------------------------------------------------------
__global__ void __launch_bounds__(256, 1)
cbow_gemm_stats_kernel(const float* __restrict__ hidden,
                       const float* __restrict__ W2,
                       float* __restrict__ logits,
                       float* __restrict__ stats) {
    __shared__ float sh[16 * 132];          // hidden tile, padded stride 132
    __shared__ float sM[WPB * 16];
    __shared__ float sS[WPB * 16];

    const int tid  = threadIdx.x;
    const int wave = tid >> 5;
    const int lane = tid & 31;
    const int rb   = blockIdx.x / SEG;      // row-tile index
    const int seg  = blockIdx.x % SEG;      // column segment
    const int rowBase = rb * 16;

    // cooperative load of hidden tile [16][128] into padded LDS
    for (int i = tid; i < 16 * EMBED; i += blockDim.x) {
        int r = i >> 7, k = i & 127;
        sh[r * 132 + k] = hidden[(size_t)(rowBase + r) * EMBED + k];
    }
    __syncthreads();

    const int n16   = lane & 15;            // N / M position within half-wave
    const int khalf = (lane >> 4) * 2;      // k-slot offset for upper half
    const int rhalf = (lane >> 4) * 8;      // C rows 0-7 vs 8-15

    // A fragments: hidden rows striped per ISA 16x4 f32 A layout, all K in regs
    v2f afrag[32];
#pragma unroll
    for (int t = 0; t < 32; ++t) {
        afrag[t] = *(const v2f*)&sh[n16 * 132 + t * 4 + khalf];
    }

    float m[8], s[8];
#pragma unroll
    for (int i = 0; i < 8; ++i) { m[i] = NEG_HUGE; s[i] = 0.0f; }

    const int gw = seg * WPB + wave;        // global wave slot over segments

    // prefetch the first tile this wave will touch (one line per K-row)
    if (gw < NTILE) {
        const float* pf = W2 + (size_t)lane * VOCAB + gw * 16;
        __builtin_prefetch(pf, 0, 3);
        __builtin_prefetch(pf + (size_t)32 * VOCAB, 0, 3);
        __builtin_prefetch(pf + (size_t)64 * VOCAB, 0, 3);
        __builtin_prefetch(pf + (size_t)96 * VOCAB, 0, 3);
    }

    for (int ct = gw; ct < NTILE; ct += TSTRIDE) {
        const int col = ct * 16 + n16;

        // prefetch next tile (cacheline-per-K-row; tile start is 64B aligned)
        const int nt = ct + TSTRIDE;
        if (nt < NTILE) {
            const float* pf = W2 + (size_t)lane * VOCAB + nt * 16;
            __builtin_prefetch(pf, 0, 3);
            __builtin_prefetch(pf + (size_t)32 * VOCAB, 0, 3);
            __builtin_prefetch(pf + (size_t)64 * VOCAB, 0, 3);
            __builtin_prefetch(pf + (size_t)96 * VOCAB, 0, 3);
        }

        v8f c = {};
        // two K-chunks of 16 WMMAs each: caps live B registers at 32/chunk
#pragma unroll
        for (int half = 0; half < 2; ++half) {
            v2f b[16];
            const float* bp = W2 + ((size_t)(half * 64 + khalf)) * VOCAB + col;
#pragma unroll
            for (int t = 0; t < 16; ++t) {
                b[t].x = bp[0];
                b[t].y = bp[VOCAB];
                bp += (size_t)4 * VOCAB;
            }
#pragma unroll
            for (int t = 0; t < 16; ++t) {
                c = __builtin_amdgcn_wmma_f32_16x16x4_f32(
                        false, afrag[half * 16 + t], false, b[t], (short)0, c,
                        false, false);
            }
        }

        float* lrow = logits + (size_t)(rowBase + rhalf) * VOCAB + col;
#pragma unroll
        for (int i = 0; i < 8; ++i) {
            float v = c[i];
            lrow[(size_t)i * VOCAB] = v;
            // online (max, sum-of-exp) update, per lane over its own columns
            if (v > m[i]) {
                s[i] = s[i] * __expf(m[i] - v) + 1.0f;
                m[i] = v;
            } else {
                s[i] += __expf(v - m[i]);
            }
        }
    }

    // combine across the 16 lanes of each half-wave
#pragma unroll
    for (int i = 0; i < 8; ++i) {
#pragma unroll
        for (int mask = 1; mask < 16; mask <<= 1) {
            float om = __shfl_xor(m[i], mask, 32);
            float os = __shfl_xor(s[i], mask, 32);
            float M  = fmaxf(m[i], om);
            s[i] = s[i] * __expf(m[i] - M) + os * __expf(om - M);
            m[i] = M;
        }
    }
    if (lane == 0 || lane == 16) {
#pragma unroll
        for (int i = 0; i < 8; ++i) {
            sM[wave * 16 + rhalf + i] = m[i];
            sS[wave * 16 + rhalf + i] = s[i];
        }
    }
    __syncthreads();

    // combine across waves, write per-segment stats
    if (tid < 16) {
        float M = NEG_HUGE;
#pragma unroll
        for (int w = 0; w < WPB; ++w) M = fmaxf(M, sM[w * 16 + tid]);
        float S = 0.0f;
#pragma unroll
        for (int w = 0; w < WPB; ++w) S += sS[w * 16 + tid] * __expf(sM[w * 16 + tid] - M);
        size_t sidx = ((size_t)(rowBase + tid) * SEG + seg) * 2;
        stats[sidx + 0] = M;
        stats[sidx + 1] = S;
    }
}

// ---------------------------------------------------------------------------
// Kernel 3: softmax normalize: out2 = exp(logits - M_row) / S_row
// grid = BATCH blocks, one row per block, float4 streaming
// ---------------------------------------------------------------------------
__global__ void cbow_normalize_kernel(const float* __restrict__ logits,
                                      const float* __restrict__ stats,
                                      float* __restrict__ out2) {
    const int row = blockIdx.x;

    // fold the SEG per-segment (m, s) pairs into final (M, 1/S)
    float M = NEG_HUGE;
    float sm[SEG], ss[SEG];
#pragma unroll
    for (int g = 0; g < SEG; ++g) {
        sm[g] = stats[((size_t)row * SEG + g) * 2 + 0];
        ss[g] = stats[((size_t)row * SEG + g) * 2 + 1];
        M = fmaxf(M, sm[g]);
    }
    float S = 0.0f;
#pragma unroll
    for (int g = 0; g < SEG; ++g) S += ss[g] * __expf(sm[g] - M);
    const float invS = 1.0f / S;

    const float4* src = (const float4*)(logits + (size_t)row * VOCAB);
    float4*       dst = (float4*)(out2 + (size_t)row * VOCAB);
    for (int j = threadIdx.x; j < VOCAB / 4; j += blockDim.x) {
        float4 l = src[j];
        float4 r;
        r.x = __expf(l.x - M) * invS;
        r.y = __expf(l.y - M) * invS;
        r.z = __expf(l.z - M) * invS;
        r.w = __expf(l.w - M) * invS;
        dst[j] = r;
    }
}

// ---------------------------------------------------------------------------
extern "C" void kernel_launch(void* const* d_in, const int* in_sizes, int n_in,
                              void* d_out, int out_size, void* d_ws, size_t ws_size,
                              hipStream_t stream) {
    const int*   inputs = (const int*)d_in[0];
    const float* W1     = (const float*)d_in[1];
    const float* W2     = (const float*)d_in[2];

    float* logits  = (float*)d_out;                          // [2048][100000]
    float* softmax = (float*)d_out + (size_t)BATCH * VOCAB;  // [2048][100000]

    float* hidden = (float*)d_ws;                            // 2048*128 floats
    float* stats  = hidden + (size_t)BATCH * EMBED;          // 2048*SEG*2 floats

    // 1) hidden = mean of gathered embeddings
    cbow_hidden_kernel<<<(BATCH * EMBED) / 256, 256, 0, stream>>>(inputs, W1, hidden);

    // 2) WMMA GEMM + logits store + online softmax stats (per column segment)
    cbow_gemm_stats_kernel<<<(BATCH / 16) * SEG, 256, 0, stream>>>(hidden, W2, logits, stats);

    // 3) normalize into softmax output
    cbow_normalize_kernel<<<BATCH, 256, 0, stream>>>(logits, stats, softmax);
}